// QuantumEncoder_84885733638565
// MI455X (gfx1250) — compile-verified
//
#include <hip/hip_runtime.h>

typedef float v2f __attribute__((ext_vector_type(2)));
typedef float v8f __attribute__((ext_vector_type(8)));

#define QDIM   256
#define QBATCH 32768

// ---------------------------------------------------------------------------
// Kernel 1: build the total circuit matrix M = QFT * G1^T * G2^T * ...
// Each row of M transforms independently, exactly like a state vector.
// 8 threads cooperate on one row held in LDS. Output stored transposed
// (MT[n*256 + k] = M[k][n]) so the GEMM B-loads are contiguous in K.
// ---------------------------------------------------------------------------
__global__ void qenc_build(const float* __restrict__ wrot,
                           const float* __restrict__ wrz,
                           float* __restrict__ MrT,
                           float* __restrict__ MiT) {
    extern __shared__ float lds[];
    const int tid = threadIdx.x;
    const int rl  = tid >> 3;          // local row 0..31
    const int u   = tid & 7;           // sub-thread 0..7 within row
    const int r   = blockIdx.x * 32 + rl;   // global row (= k index of M)

    float* sre = lds + rl * 514;       // 257-float stride per plane: no bank conflicts
    float* sim = sre + 257;

    // init: row r of QFT: exp(2*pi*i*r*k/256)/16
    const float w = 6.28318530717958647692f / 256.0f;
    for (int j = 0; j < 32; ++j) {
        int k  = u * 32 + j;
        int ph = (r * k) & 255;
        float sv, cv;
        sincosf(w * (float)ph, &sv, &cv);
        sre[k] = cv * 0.0625f;
        sim[k] = sv * 0.0625f;
    }
    __syncthreads();

    for (int l = 0; l < 4; ++l) {
        // single-qubit fused RZ(ph)*RY(th) gates
        for (int q = 0; q < 8; ++q) {
            float th  = wrot[l * 8 + q];
            float phz = wrz[l * 8 + q];
            float c = cosf(0.5f * th), s = sinf(0.5f * th);
            float emr = cosf(0.5f * phz), emi = -sinf(0.5f * phz); // e^{-i ph/2}
            // U = [[em*c, -em*s], [ep*s, ep*c]],  ep = conj(em)
            float U00r =  emr * c, U00i =  emi * c;
            float U01r = -emr * s, U01i = -emi * s;
            float U10r =  emr * s, U10i = -emi * s;
            float U11r =  emr * c, U11i = -emi * c;
            int bit = 7 - q, mask = 1 << bit;
            for (int j = 0; j < 16; ++j) {
                int p  = u * 16 + j;
                int k0 = ((p >> bit) << (bit + 1)) | (p & (mask - 1));
                int k1 = k0 | mask;
                float ar = sre[k0], ai = sim[k0];
                float br = sre[k1], bi = sim[k1];
                sre[k0] = U00r * ar - U00i * ai + U01r * br - U01i * bi;
                sim[k0] = U00r * ai + U00i * ar + U01r * bi + U01i * br;
                sre[k1] = U10r * ar - U10i * ai + U11r * br - U11i * bi;
                sim[k1] = U10r * ai + U10i * ar + U11r * bi + U11i * br;
            }
            __syncthreads();
        }
        // CNOT ring: control q, target (q+1)%8
        for (int q = 0; q < 8; ++q) {
            int cb = 7 - q, tb = 7 - ((q + 1) & 7);
            int tmask = 1 << tb;
            for (int j = 0; j < 32; ++j) {
                int k = u * 32 + j;
                if (((k >> cb) & 1) && !((k >> tb) & 1)) {
                    int k2 = k | tmask;
                    float tr = sre[k], ti = sim[k];
                    sre[k]  = sre[k2]; sim[k]  = sim[k2];
                    sre[k2] = tr;      sim[k2] = ti;
                }
            }
            __syncthreads();
        }
    }

    // store transposed: MT[n][k=r]
    for (int j = 0; j < 32; ++j) {
        int n = u * 32 + j;
        MrT[n * 256 + r] = sre[n];
        MiT[n * 256 + r] = sim[n];
    }
}

// ---------------------------------------------------------------------------
// Kernel 2: out[b,n] = |(X @ M)[b,n]|^2 / ||X[b]||^2 via f32 WMMA.
// One wave computes a 32(M) x 32(N) tile: 2 row-groups x 2 col-groups x
// (Re,Im) = 8 accumulator chains. 6 b64 loads feed 8 WMMAs per K-step of 4.
// ---------------------------------------------------------------------------
#define WMMA_F32(A, B, C) \
    __builtin_amdgcn_wmma_f32_16x16x4_f32(false, (A), false, (B), (short)0, (C), false, false)

__global__ void qenc_gemm(const float* __restrict__ X,
                          const float* __restrict__ MrT,
                          const float* __restrict__ MiT,
                          float* __restrict__ out) {
    const int lane = threadIdx.x & 31;
    const int wid  = blockIdx.x * 8 + (threadIdx.x >> 5); // 0..8191
    const int n0   = (wid & 7) * 32;                      // 8 N-blocks of 32
    const int m0   = (wid >> 3) * 32;                     // 1024 M-blocks of 32
    const int half = lane >> 4;   // 0|1
    const int l15  = lane & 15;

    // A 16x4 layout: lane<16 holds (K0,K1) of row l15, lane>=16 holds (K2,K3).
    // B 4x16 layout mirrors it with columns. Both are contiguous float2 loads.
    const float* ap0  = X   + (m0 + l15) * 256 + 2 * half;
    const float* ap1  = ap0 + 16 * 256;
    const float* brp0 = MrT + (n0 + l15) * 256 + 2 * half;
    const float* brp1 = brp0 + 16 * 256;
    const float* bip0 = MiT + (n0 + l15) * 256 + 2 * half;
    const float* bip1 = bip0 + 16 * 256;

    v8f accR00 = {}, accI00 = {}, accR01 = {}, accI01 = {};
    v8f accR10 = {}, accI10 = {}, accR11 = {}, accI11 = {};
    float nrm0 = 0.0f, nrm1 = 0.0f;

#pragma unroll 2
    for (int k = 0; k < 256; k += 4) {
        v2f A0  = *(const v2f*)(ap0 + k);
        v2f A1  = *(const v2f*)(ap1 + k);
        v2f Br0 = *(const v2f*)(brp0 + k);
        v2f Br1 = *(const v2f*)(brp1 + k);
        v2f Bi0 = *(const v2f*)(bip0 + k);
        v2f Bi1 = *(const v2f*)(bip1 + k);
        nrm0 += A0.x * A0.x + A0.y * A0.y;  // lanes L and L+16 jointly cover all K
        nrm1 += A1.x * A1.x + A1.y * A1.y;
        accR00 = WMMA_F32(A0, Br0, accR00);
        accI00 = WMMA_F32(A0, Bi0, accI00);
        accR01 = WMMA_F32(A0, Br1, accR01);
        accI01 = WMMA_F32(A0, Bi1, accI01);
        accR10 = WMMA_F32(A1, Br0, accR10);
        accI10 = WMMA_F32(A1, Bi0, accI10);
        accR11 = WMMA_F32(A1, Br1, accR11);
        accI11 = WMMA_F32(A1, Bi1, accI11);
    }

    // row norms: lanes L and L+16 hold complementary K partials of row (L&15)
    nrm0 += __shfl_xor(nrm0, 16, 32);   // lane j holds ||x[m0      + (j&15)]||^2
    nrm1 += __shfl_xor(nrm1, 16, 32);   // lane j holds ||x[m0 + 16 + (j&15)]||^2

    // C/D layout: VGPR i -> row (i + 8*half), col = l15 (per 16x16 sub-tile)
#pragma unroll
    for (int i = 0; i < 8; ++i) {
        int row    = i + 8 * half;
        float inv0 = 1.0f / __shfl(nrm0, row, 32);
        float inv1 = 1.0f / __shfl(nrm1, row, 32);
        float* o0  = out + (m0 + row) * 256 + n0 + l15;
        float* o1  = out + (m0 + 16 + row) * 256 + n0 + l15;
        o0[0]  = (accR00[i] * accR00[i] + accI00[i] * accI00[i]) * inv0;
        o0[16] = (accR01[i] * accR01[i] + accI01[i] * accI01[i]) * inv0;
        o1[0]  = (accR10[i] * accR10[i] + accI10[i] * accI10[i]) * inv1;
        o1[16] = (accR11[i] * accR11[i] + accI11[i] * accI11[i]) * inv1;
    }
}

// ---------------------------------------------------------------------------
extern "C" void kernel_launch(void* const* d_in, const int* in_sizes, int n_in,
                              void* d_out, int out_size, void* d_ws, size_t ws_size,
                              hipStream_t stream) {
    const float* x    = (const float*)d_in[0];
    const float* wrot = (const float*)d_in[1];
    const float* wrz  = (const float*)d_in[2];
    float* out = (float*)d_out;

    float* MrT = (float*)d_ws;              // 256*256 f32 = 256 KB
    float* MiT = MrT + QDIM * QDIM;         // next 256 KB

    size_t lds_bytes = 32 * 514 * sizeof(float);  // 32 rows * (2*257) floats
    qenc_build<<<8, 256, lds_bytes, stream>>>(wrot, wrz, MrT, MiT);

    // (32768/32) M-blocks x (256/32) N-blocks = 8192 waves, 8 waves/block
    qenc_gemm<<<1024, 256, 0, stream>>>(x, MrT, MiT, out);
}